// ElasticTransform_27693949124764
// MI455X (gfx1250) — compile-verified
//
#include <hip/hip_runtime.h>
#include <hip/hip_bf16.h>

typedef __attribute__((ext_vector_type(2))) float v2f;
typedef __attribute__((ext_vector_type(8))) float v8f;

#define HH 2048
#define WW 2048
#define HWSZ (HH * WW)
#define KS 97
#define PAD 48
#define NCHUNK 28   // K padded to 112 = 28 * 4

// ---------------------------------------------------------------------------
// Kernel 0: runtime Gaussian weights + alphas.
// wbuf layout: [0..127]=w_ch0, [128..255]=w_ch1, [256]=alpha0, [257]=alpha1
// ---------------------------------------------------------------------------
__global__ __launch_bounds__(256) void et_weights_kernel(
    const float* __restrict__ log_sigma, const float* __restrict__ log_alpha,
    float* __restrict__ wbuf) {
  __shared__ float g[2][128];
  const int t  = threadIdx.x;
  const int ch = t >> 7;
  const int i  = t & 127;
  const float sigma = expf(log_sigma[ch]);
  float val = 0.0f;
  if (i < KS) {
    const float xx = (float)(i - PAD);
    val = expf(-xx * xx / (2.0f * sigma * sigma));
  }
  g[ch][i] = val;
  __syncthreads();
  float s = 0.0f;
  for (int j = 0; j < KS; ++j) s += g[ch][j];
  wbuf[ch * 128 + i] = val / s;
  if (i == 0) wbuf[256 + ch] = expf(log_alpha[ch]);
}

// ---------------------------------------------------------------------------
// Kernel 1: vertical blur of d = d_rand*2-1 (zero-padded), via WMMA f32 16x16x4.
// OUT[16x16] = M(16x112) * P(112x16), M[r][s] = w[s-r].
// Block = 128 thr (4 waves), covers 64 rows x 64 cols of one channel.
// ---------------------------------------------------------------------------
__global__ __launch_bounds__(128) void et_blur_v_kernel(
    const float* __restrict__ d_rand, const float* __restrict__ wbuf,
    float* __restrict__ tmp) {
  __shared__ float patch[160 * 64];  // rows y0-48 .. y0+111, 64 cols (40 KB)

  const int ch = blockIdx.z;
  const int x0 = blockIdx.x * 64;
  const int y0 = blockIdx.y * 64;
  const float* __restrict__ src = d_rand + (size_t)ch * HWSZ;
  const float* __restrict__ wv  = wbuf + ch * 128;

  const int tid = threadIdx.x;
  // cooperative zero-padded load, 2560 float4s, fully coalesced
  for (int i = tid; i < 160 * 16; i += 128) {
    const int row = i >> 4;
    const int c4  = (i & 15) * 4;
    const int gy  = y0 - PAD + row;
    float4 v = make_float4(0.f, 0.f, 0.f, 0.f);
    if (gy >= 0 && gy < HH) {
      v = *(const float4*)(src + (size_t)gy * WW + x0 + c4);
      v.x = v.x * 2.0f - 1.0f;
      v.y = v.y * 2.0f - 1.0f;
      v.z = v.z * 2.0f - 1.0f;
      v.w = v.w * 2.0f - 1.0f;
    }
    *(float4*)&patch[row * 64 + c4] = v;
  }

  const int lane = tid & 31;
  const int wid  = tid >> 5;        // wave id: rows 16*wid .. 16*wid+15
  const int r    = lane & 15;       // M row (A) / N col (B,D)
  const int khi  = (lane >> 4) * 2; // K offset for upper half-wave

  // A = weight matrix chunks, resident in registers (reused for 4 sub-tiles)
  v2f aw[NCHUNK];
#pragma unroll
  for (int t = 0; t < NCHUNK; ++t) {
#pragma unroll
    for (int v = 0; v < 2; ++v) {
      const int s   = 4 * t + khi + v;
      const int idx = s - r;
      aw[t][v] = (idx >= 0 && idx < KS) ? wv[idx] : 0.0f;
    }
  }

  __syncthreads();

  const v8f zero = {0.f, 0.f, 0.f, 0.f, 0.f, 0.f, 0.f, 0.f};
  v8f acc[4];
#pragma unroll
  for (int c = 0; c < 4; ++c) acc[c] = zero;

#pragma unroll
  for (int t = 0; t < NCHUNK; ++t) {
    const int rowb = 16 * wid + 4 * t + khi;  // patch row of B[k0]
#pragma unroll
    for (int c = 0; c < 4; ++c) {
      v2f b;
      b[0] = patch[rowb * 64 + 16 * c + r];
      b[1] = patch[(rowb + 1) * 64 + 16 * c + r];
      acc[c] = __builtin_amdgcn_wmma_f32_16x16x4_f32(
          false, aw[t], false, b, (short)0, acc[c], false, false);
    }
  }

  // D layout: VGPR j, lanes0-15 -> M=j, lanes16-31 -> M=j+8
  float* __restrict__ dst = tmp + (size_t)ch * HWSZ;
#pragma unroll
  for (int c = 0; c < 4; ++c) {
#pragma unroll
    for (int j = 0; j < 8; ++j) {
      const int row = y0 + 16 * wid + j + 8 * (lane >> 4);
      dst[(size_t)row * WW + x0 + 16 * c + r] = acc[c][j];
    }
  }
}

// ---------------------------------------------------------------------------
// Kernel 2: horizontal blur of tmp (zero-padded) * alpha -> disp.
// OUT[16x16] = P(16x112) * M'(112x16), M'[s][c] = w[s-c].
// ---------------------------------------------------------------------------
__global__ __launch_bounds__(128) void et_blur_h_kernel(
    const float* __restrict__ tmp, const float* __restrict__ wbuf,
    float* __restrict__ disp) {
  __shared__ float patch[64 * 160];  // 64 rows, cols x0-48 .. x0+111 (40 KB)

  const int ch = blockIdx.z;
  const int x0 = blockIdx.x * 64;
  const int y0 = blockIdx.y * 64;
  const float* __restrict__ src = tmp + (size_t)ch * HWSZ;
  const float* __restrict__ wv  = wbuf + ch * 128;
  const float alpha = wbuf[256 + ch];

  const int tid = threadIdx.x;
  for (int i = tid; i < 64 * 40; i += 128) {
    const int row = i / 40;
    const int c4  = (i % 40) * 4;
    const int gx  = x0 - PAD + c4;   // multiple of 4 -> float4 fully in or out
    float4 v = make_float4(0.f, 0.f, 0.f, 0.f);
    if (gx >= 0 && gx < WW) {
      v = *(const float4*)(src + (size_t)(y0 + row) * WW + gx);
    }
    *(float4*)&patch[row * 160 + c4] = v;
  }

  const int lane = tid & 31;
  const int wid  = tid >> 5;
  const int r    = lane & 15;
  const int khi  = (lane >> 4) * 2;

  // B = transposed weight matrix chunks in registers: B[k][c] = w[4t+k-c]
  v2f bw[NCHUNK];
#pragma unroll
  for (int t = 0; t < NCHUNK; ++t) {
#pragma unroll
    for (int v = 0; v < 2; ++v) {
      const int idx = 4 * t + khi + v - r;
      bw[t][v] = (idx >= 0 && idx < KS) ? wv[idx] : 0.0f;
    }
  }

  __syncthreads();

  const v8f zero = {0.f, 0.f, 0.f, 0.f, 0.f, 0.f, 0.f, 0.f};
  v8f acc[4];
#pragma unroll
  for (int c = 0; c < 4; ++c) acc[c] = zero;

#pragma unroll
  for (int t = 0; t < NCHUNK; ++t) {
#pragma unroll
    for (int c = 0; c < 4; ++c) {
      // A[r][k] = patch[16*wid + r][16*c + 4t + k], k consecutive -> float2 LDS read
      const int col = 16 * c + 4 * t + khi;   // even -> 8B aligned
      const v2f a = *(const v2f*)&patch[(16 * wid + r) * 160 + col];
      acc[c] = __builtin_amdgcn_wmma_f32_16x16x4_f32(
          false, a, false, bw[t], (short)0, acc[c], false, false);
    }
  }

  float* __restrict__ dst = disp + (size_t)ch * HWSZ;
#pragma unroll
  for (int c = 0; c < 4; ++c) {
#pragma unroll
    for (int j = 0; j < 8; ++j) {
      const int row = y0 + 16 * wid + j + 8 * (lane >> 4);
      dst[(size_t)row * WW + x0 + 16 * c + r] = acc[c][j] * alpha;
    }
  }
}

// ---------------------------------------------------------------------------
// Kernel 3: bilinear warp of image (3ch) + mask (1ch).
// Reference quirk: disp channel 0 displaces x, channel 1 displaces y.
// ---------------------------------------------------------------------------
__global__ __launch_bounds__(256) void et_warp_kernel(
    const float* __restrict__ image, const float* __restrict__ mask,
    const float* __restrict__ disp, float* __restrict__ out) {
  const int idx = blockIdx.x * 256 + threadIdx.x;
  if (idx >= HWSZ) return;
  const int x = idx & (WW - 1);
  const int y = idx >> 11;

  const float dx_disp = disp[idx];          // ch0 -> x
  const float dy_disp = disp[HWSZ + idx];   // ch1 -> y

  const float gx = -1.0f + 2.0f * (float)x / (float)(WW - 1);
  const float gy = -1.0f + 2.0f * (float)y / (float)(HH - 1);
  const float sx = fminf(1.0f, fmaxf(-1.0f, gx + dx_disp));
  const float sy = fminf(1.0f, fmaxf(-1.0f, gy + dy_disp));

  const float fx = (sx + 1.0f) * 0.5f * (float)(WW - 1);
  const float fy = (sy + 1.0f) * 0.5f * (float)(HH - 1);
  const float x0f = floorf(fx);
  const float y0f = floorf(fy);
  const float wx = fx - x0f;
  const float wy = fy - y0f;

  const int x0i = min(max((int)x0f, 0), WW - 1);
  const int x1i = min(max((int)x0f + 1, 0), WW - 1);
  const int y0i = min(max((int)y0f, 0), HH - 1);
  const int y1i = min(max((int)y0f + 1, 0), HH - 1);

  const size_t i00 = (size_t)y0i * WW + x0i;
  const size_t i01 = (size_t)y0i * WW + x1i;
  const size_t i10 = (size_t)y1i * WW + x0i;
  const size_t i11 = (size_t)y1i * WW + x1i;

#pragma unroll
  for (int c = 0; c < 3; ++c) {
    const float* p = image + (size_t)c * HWSZ;
    const float top = p[i00] * (1.0f - wx) + p[i01] * wx;
    const float bot = p[i10] * (1.0f - wx) + p[i11] * wx;
    out[(size_t)c * HWSZ + idx] = top * (1.0f - wy) + bot * wy;
  }
  {
    const float top = mask[i00] * (1.0f - wx) + mask[i01] * wx;
    const float bot = mask[i10] * (1.0f - wx) + mask[i11] * wx;
    out[(size_t)3 * HWSZ + idx] = top * (1.0f - wy) + bot * wy;
  }
}

// ---------------------------------------------------------------------------
extern "C" void kernel_launch(void* const* d_in, const int* in_sizes, int n_in,
                              void* d_out, int out_size, void* d_ws, size_t ws_size,
                              hipStream_t stream) {
  (void)in_sizes; (void)n_in; (void)out_size; (void)ws_size;
  const float* image     = (const float*)d_in[0];
  const float* mask      = (const float*)d_in[1];
  const float* d_rand    = (const float*)d_in[2];
  const float* log_sigma = (const float*)d_in[3];
  const float* log_alpha = (const float*)d_in[4];
  float* out  = (float*)d_out;

  float* tmp  = (float*)d_ws;              // 2 * H * W  (vertical-blurred)
  float* disp = tmp + (size_t)2 * HWSZ;    // 2 * H * W  (final displacement)
  float* wbuf = tmp + (size_t)4 * HWSZ;    // 258 floats (weights + alphas)

  et_weights_kernel<<<1, 256, 0, stream>>>(log_sigma, log_alpha, wbuf);
  et_blur_v_kernel<<<dim3(WW / 64, HH / 64, 2), 128, 0, stream>>>(d_rand, wbuf, tmp);
  et_blur_h_kernel<<<dim3(WW / 64, HH / 64, 2), 128, 0, stream>>>(tmp, wbuf, disp);
  et_warp_kernel<<<(HWSZ + 255) / 256, 256, 0, stream>>>(image, mask, disp, out);
}